// DRCapsule_50311246905740
// MI455X (gfx1250) — compile-verified
//
#include <hip/hip_runtime.h>
#include <hip/hip_bf16.h>

typedef float v2f __attribute__((ext_vector_type(2)));
typedef float v8f __attribute__((ext_vector_type(8)));

#define NB   128            // batch
#define NI   2048           // input capsules
#define NU   16             // input units (GEMM K)
#define NJ   32             // output capsules
#define NV   16             // output units
#define NJV  512            // J*V (GEMM N)
#define GEMM_BLOCKS 256
#define I_PER_GBLK (NI / GEMM_BLOCKS)   // 8
#define RCHUNKS 32
#define I_PER_RBLK (NI / RCHUNKS)       // 64
#define EPSQ 1e-9f

// ---------------------------------------------------------------------------
// Kernel 1: votes u[b,i,j,v] = sum_u x[b,i,u] * W[i,j,v,u] via v_wmma_f32_16x16x4_f32.
// Block = 8 waves (wave w -> batch rows 16w..16w+15), chunk of 8 capsules,
// all 512 output columns. nt (j-tile) is the outer, non-unrolled loop; the 8
// capsules are fully unrolled inside -> 8 independent 4-deep WMMA chains.
// Round-0 fusion: with nt outer, each (b,col) partial s0 is produced by one
// wave in one nt iteration, so it accumulates in v8f registers over the 8
// capsules and stores straight to global (no LDS, no barriers, no atomics).
// u is a 537MB single-use stream -> non-temporal stores; x/W are single-use
// -> non-temporal loads; part0 stays RT so the following squash hits L2.
// ---------------------------------------------------------------------------
__global__ __launch_bounds__(256) void caps_gemm_wmma(
    const float* __restrict__ x, const float* __restrict__ W,
    float* __restrict__ u, float* __restrict__ part0) {
  const int t    = threadIdx.x;
  const int wid  = t >> 5;       // wave id 0..7 -> batch tile (16 rows)
  const int lane = t & 31;
  const int row  = lane & 15;    // M row within tile / N col within tile
  const int hi   = lane >> 4;    // 0: K{0,1}  1: K{2,3}  (f32 A/B VGPR layout)
  const int b0   = wid * 16;
  const int i0   = blockIdx.x * I_PER_GBLK;

  // A fragments for all 8 capsules of this chunk: x[b0+row, i0+ii, 0:16]
  v2f A[I_PER_GBLK][4];
  #pragma unroll
  for (int ii = 0; ii < I_PER_GBLK; ++ii) {
    const v2f* xp = (const v2f*)(x + ((size_t)(b0 + row) * NI + (i0 + ii)) * NU + 2 * hi);
    A[ii][0] = __builtin_nontemporal_load(xp + 0);
    A[ii][1] = __builtin_nontemporal_load(xp + 2);
    A[ii][2] = __builtin_nontemporal_load(xp + 4);
    A[ii][3] = __builtin_nontemporal_load(xp + 6);
  }

  // Per-row u store offsets (elements); ii/nt contributions fold to immediates
  // or cheap bumps. Storing via the global base pointer keeps global_store.
  size_t ubO[8];
  #pragma unroll
  for (int r = 0; r < 8; ++r)
    ubO[r] = ((size_t)(b0 + hi * 8 + r) * NI + i0) * NJV + row;
  size_t pbO = (size_t)blockIdx.x * (NB * NJV) + (b0 + hi * 8) * NJV + row;
  const float* wptr = W + (size_t)i0 * (NJV * NU) + row * NU + 2 * hi;

  #pragma unroll 1
  for (int nt = 0; nt < NJ; ++nt) {
    v8f sacc = {};
    #pragma unroll
    for (int ii = 0; ii < I_PER_GBLK; ++ii) {
      const v2f* bp = (const v2f*)(wptr + ii * (NJV * NU));  // B[k][col] = W[i, col, k]
      const v2f w0 = __builtin_nontemporal_load(bp + 0);
      const v2f w1 = __builtin_nontemporal_load(bp + 2);
      const v2f w2 = __builtin_nontemporal_load(bp + 4);
      const v2f w3 = __builtin_nontemporal_load(bp + 6);

      v8f acc = {};
      acc = __builtin_amdgcn_wmma_f32_16x16x4_f32(false, A[ii][0], false, w0, (short)0, acc, false, false);
      acc = __builtin_amdgcn_wmma_f32_16x16x4_f32(false, A[ii][1], false, w1, (short)0, acc, false, false);
      acc = __builtin_amdgcn_wmma_f32_16x16x4_f32(false, A[ii][2], false, w2, (short)0, acc, false, false);
      acc = __builtin_amdgcn_wmma_f32_16x16x4_f32(false, A[ii][3], false, w3, (short)0, acc, false, false);

      #pragma unroll
      for (int r = 0; r < 8; ++r)                 // C/D: M = r (+8 for hi lanes)
        __builtin_nontemporal_store(acc[r], u + ubO[r] + ii * NJV);
      sacc += acc;
    }
    #pragma unroll
    for (int r = 0; r < 8; ++r)
      part0[pbO + r * NJV] = sacc[r] * (1.0f / (float)NJ);

    #pragma unroll
    for (int r = 0; r < 8; ++r) ubO[r] += 16;     // next j-tile column block
    pbO += 16;
    wptr += 16 * NU;
  }
}

// ---------------------------------------------------------------------------
// Routing round (r>=1). One block = (batch b, chunk of 64 capsules i).
// Per i: logits l_j = prev + sum_v u[b,i,j,v]*vprev[b,j,v]; softmax over j (one
// wave32 shuffle reduction); s_loc[n] += c_{n/16} * u[n]. Deterministic: each
// block writes its own s-partial slice (no atomics). u loads are non-temporal
// (single-use 537MB stream).
// ---------------------------------------------------------------------------
__global__ __launch_bounds__(256) void caps_route(
    const float* __restrict__ u, const float* __restrict__ lin,
    float* __restrict__ lout, const float* __restrict__ vprev,
    float* __restrict__ spart, int hasLin, int writeL) {
  __shared__ float prod[NJV];
  __shared__ float cj[NJ];
  __shared__ float sloc[NJV];
  __shared__ float vp[NJV];
  const int t = threadIdx.x;
  const int b = blockIdx.x >> 5;
  const int chunk = blockIdx.x & 31;
  const int i0 = chunk * I_PER_RBLK;

  sloc[t] = 0.0f; sloc[t + 256] = 0.0f;
  vp[t] = vprev[b * NJV + t];
  vp[t + 256] = vprev[b * NJV + t + 256];
  __syncthreads();

  for (int ii = 0; ii < I_PER_RBLK; ++ii) {
    const int i = i0 + ii;
    const float* up = u + ((size_t)b * NI + i) * NJV;
    const float u0 = __builtin_nontemporal_load(up + t);
    const float u1 = __builtin_nontemporal_load(up + t + 256);
    prod[t] = u0 * vp[t];
    prod[t + 256] = u1 * vp[t + 256];
    __syncthreads();

    if (t < NJ) {                                   // one full wave32: j = lane
      float l = hasLin ? lin[((size_t)b * NI + i) * NJ + t] : 0.0f;
      #pragma unroll
      for (int v = 0; v < NV; ++v) l += prod[t * NV + v];
      if (writeL) lout[((size_t)b * NI + i) * NJ + t] = l;
      float m = l;
      #pragma unroll
      for (int off = 16; off; off >>= 1) m = fmaxf(m, __shfl_xor(m, off, 32));
      const float e = __expf(l - m);
      float s = e;
      #pragma unroll
      for (int off = 16; off; off >>= 1) s += __shfl_xor(s, off, 32);
      cj[t] = e / s;
    }
    __syncthreads();

    sloc[t]       += cj[t >> 4] * u0;
    sloc[t + 256] += cj[(t + 256) >> 4] * u1;
    __syncthreads();
  }

  float* sp = spart + ((size_t)b * RCHUNKS + chunk) * NJV;
  sp[t] = sloc[t];
  sp[t + 256] = sloc[t + 256];
}

// ---------------------------------------------------------------------------
// Reduce s-partials (fixed order -> deterministic), add bias, squash.
// 512 threads = one n each; ||s_j||^2 via width-16 shuffle reduction.
// ---------------------------------------------------------------------------
__global__ __launch_bounds__(512) void caps_squash(
    const float* __restrict__ part, const float* __restrict__ bias,
    float* __restrict__ vout, int nparts, int pstride, int bstride) {
  const int b = blockIdx.x;
  const int t = threadIdx.x;             // n = j*16 + v
  float s = 0.0f;
  for (int p = 0; p < nparts; ++p)
    s += part[(size_t)p * pstride + (size_t)b * bstride + t];
  s += bias[t];
  float sq = s * s;
  #pragma unroll
  for (int off = 8; off; off >>= 1) sq += __shfl_xor(sq, off, 16);
  vout[(size_t)b * NJV + t] = s * sq / (1.0f + sq) * rsqrtf(sq + EPSQ);
}

// ---------------------------------------------------------------------------
extern "C" void kernel_launch(void* const* d_in, const int* in_sizes, int n_in,
                              void* d_out, int out_size, void* d_ws, size_t ws_size,
                              hipStream_t stream) {
  const float* x    = (const float*)d_in[0];   // [128,2048,16]
  const float* W    = (const float*)d_in[1];   // [2048,32,16,16]
  const float* bias = (const float*)d_in[2];   // [32,16]
  float* out = (float*)d_out;                  // [128,32,16]
  float* ws  = (float*)d_ws;

  float* u     = ws;                                        // 134,217,728 f
  float* part0 = u     + (size_t)NB * NI * NJV;             //  16,777,216 f
  float* l1    = part0 + (size_t)GEMM_BLOCKS * NB * NJV;    //   8,388,608 f
  float* spart = l1    + (size_t)NB * NI * NJ;              //   2,097,152 f
  float* v0    = spart + (size_t)NB * RCHUNKS * NJV;        //      65,536 f
  float* v1    = v0    + (size_t)NB * NJV;                  //      65,536 f

  // Votes + fused round-0 uniform-coupling partial sums.
  caps_gemm_wmma<<<GEMM_BLOCKS, 256, 0, stream>>>(x, W, u, part0);
  // v0 = squash((1/J)*sum_i u + bias)
  caps_squash<<<NB, NJV, 0, stream>>>(part0, bias, v0, GEMM_BLOCKS, NB * NJV, NJV);

  // Round 1: logits l1 = u.v0 ; c = softmax_j(l1) ; s1 = sum_i c*u
  caps_route<<<NB * RCHUNKS, 256, 0, stream>>>(u, nullptr, l1, v0, spart, 0, 1);
  caps_squash<<<NB, NJV, 0, stream>>>(spart, bias, v1, RCHUNKS, NJV, RCHUNKS * NJV);

  // Round 2 (final): logits = l1 + u.v1 ; s2 = sum_i softmax*u ; out = squash
  caps_route<<<NB * RCHUNKS, 256, 0, stream>>>(u, l1, nullptr, v1, spart, 1, 0);
  caps_squash<<<NB, NJV, 0, stream>>>(spart, bias, out, RCHUNKS, NJV, RCHUNKS * NJV);
}